// BayesianDense_27453430956335
// MI455X (gfx1250) — compile-verified
//
#include <hip/hip_runtime.h>

#define B_ 512
#define D_ 512
#define O_ 256
#define O4_ (O_ / 4)

typedef float v2f __attribute__((ext_vector_type(2)));
typedef float v4f __attribute__((ext_vector_type(4)));
typedef float v8f __attribute__((ext_vector_type(8)));

// ---------------------------------------------------------------------------
// Kernel 1: sigma precompute.  s[d,o] = exp(0.5*W_log_sigma[d,o]) (D*O floats)
//           sb[o] = exp(0.5*b_log_sigma[o])                       (O  floats)
// ---------------------------------------------------------------------------
__global__ __launch_bounds__(256) void sigma_kernel(
    const float* __restrict__ Wls, const float* __restrict__ bls,
    float* __restrict__ s, float* __restrict__ sb) {
  int i = blockIdx.x * 256 + threadIdx.x;
  if (i < D_ * O_) s[i] = __expf(0.5f * Wls[i]);
  if (i < O_)      sb[i] = __expf(0.5f * bls[i]);
}

// ---------------------------------------------------------------------------
// Kernel 2: mean-path GEMM  out[b,o] = sum_d x[b,d] * W_mu[d,o]
// One 16x16 output tile per wave via V_WMMA_F32_16X16X4_F32, K-loop of 128.
// A 16x4 layout (ISA 7.12.2): lanes 0-15 -> (K0,K1), lanes 16-31 -> (K2,K3),
// M = lane&15.  B 4x16 symmetric.  C/D: VGPR i holds M=i (lanes 0-15) and
// M=i+8 (lanes 16-31), N = lane&15.
// ---------------------------------------------------------------------------
__global__ __launch_bounds__(256) void gemm_wmma_kernel(
    const float* __restrict__ x, const float* __restrict__ Wmu,
    float* __restrict__ out) {
  const int lane = threadIdx.x & 31;
  const int wave = threadIdx.x >> 5;
  const int tile = blockIdx.x * 8 + wave;   // 512 tiles total
  const int bm = (tile >> 4) * 16;          // batch-tile origin (32 tiles)
  const int on = (tile & 15) * 16;          // out-tile origin   (16 tiles)
  const int r  = lane & 15;
  const int h  = lane >> 4;                 // 0: K0/K1, 1: K2/K3

  v8f c = {0.f, 0.f, 0.f, 0.f, 0.f, 0.f, 0.f, 0.f};
  const float* arow = x + (bm + r) * D_;
  const float* bcol = Wmu + on + r;

  for (int k0 = 0; k0 < D_; k0 += 4) {
    const int ka = k0 + 2 * h;
    v2f a, b;
    a.x = arow[ka];
    a.y = arow[ka + 1];
    b.x = bcol[ka * O_];
    b.y = bcol[(ka + 1) * O_];
    c = __builtin_amdgcn_wmma_f32_16x16x4_f32(
        /*neg_a=*/false, a, /*neg_b=*/false, b,
        /*c_mod=*/(short)0, c, /*reuse_a=*/false, /*reuse_b=*/false);
  }

  float* orow = out + on + r;
#pragma unroll
  for (int i = 0; i < 8; ++i) {
    orow[(bm + i + 8 * h) * O_] = c[i];
  }
}

// ---------------------------------------------------------------------------
// Kernel 3: HBM-bound noise stream + bias.
// out[b,o] += sum_d x[b,d]*s[d,o]*e[b,d,o] + b_mu[o] + eb[b,o]*sb[o]
// Thread owns a float4 along o; 64 threads cover one contiguous 1KB e-row
// -> coalesced b128 loads streaming 256 MB at HBM rate.  e is read-once, so
// its loads are non-temporal (TH=NT) to keep the reused sigma/x tiles hot in
// WGP$/L2.  d unrolled x4 sharing one float4 x-load.
// ---------------------------------------------------------------------------
__device__ __forceinline__ void fma4(v4f& acc, float xs, const v4f& sv,
                                     const v4f& ev) {
  acc.x = fmaf(xs * sv.x, ev.x, acc.x);
  acc.y = fmaf(xs * sv.y, ev.y, acc.y);
  acc.z = fmaf(xs * sv.z, ev.z, acc.z);
  acc.w = fmaf(xs * sv.w, ev.w, acc.w);
}

__global__ __launch_bounds__(256) void noise_bias_kernel(
    const float* __restrict__ x, const float* __restrict__ s,
    const float* __restrict__ bmu, const float* __restrict__ sb,
    const float* __restrict__ e, const float* __restrict__ eb,
    float* __restrict__ out) {
  const int idx = blockIdx.x * 256 + threadIdx.x;  // 0 .. B*O4-1 (32768)
  const int b  = idx >> 6;   // / O4_
  const int oq = idx & 63;   // float4 slot along o

  const v4f* __restrict__ e4 = (const v4f*)e;
  const v4f* __restrict__ s4 = (const v4f*)s;
  const float* xrow = x + b * D_;
  const int ebase = b * (D_ * O4_);  // max ~16.7M float4, fits int

  v4f acc = {0.f, 0.f, 0.f, 0.f};
  for (int d = 0; d < D_; d += 4) {
    const v4f xv = *(const v4f*)(xrow + d);
    // read-once 256 MB stream: non-temporal loads (TH=NT)
    const v4f e0 = __builtin_nontemporal_load(&e4[ebase + (d + 0) * O4_ + oq]);
    const v4f e1 = __builtin_nontemporal_load(&e4[ebase + (d + 1) * O4_ + oq]);
    const v4f e2 = __builtin_nontemporal_load(&e4[ebase + (d + 2) * O4_ + oq]);
    const v4f e3 = __builtin_nontemporal_load(&e4[ebase + (d + 3) * O4_ + oq]);
    // reused sigma tiles: regular temporal loads, stay L2-resident
    const v4f s0 = s4[(d + 0) * O4_ + oq];
    const v4f s1 = s4[(d + 1) * O4_ + oq];
    const v4f s2 = s4[(d + 2) * O4_ + oq];
    const v4f s3 = s4[(d + 3) * O4_ + oq];
    fma4(acc, xv.x, s0, e0);
    fma4(acc, xv.y, s1, e1);
    fma4(acc, xv.z, s2, e2);
    fma4(acc, xv.w, s3, e3);
  }

  const v4f bm4 = ((const v4f*)bmu)[oq];
  const v4f sb4 = ((const v4f*)sb)[oq];
  const v4f eb4 = __builtin_nontemporal_load(&((const v4f*)eb)[idx]);
  v4f o = ((v4f*)out)[idx];  // contains WMMA mean-path result
  o.x += acc.x + bm4.x + eb4.x * sb4.x;
  o.y += acc.y + bm4.y + eb4.y * sb4.y;
  o.z += acc.z + bm4.z + eb4.z * sb4.z;
  o.w += acc.w + bm4.w + eb4.w * sb4.w;
  ((v4f*)out)[idx] = o;
}

// ---------------------------------------------------------------------------
extern "C" void kernel_launch(void* const* d_in, const int* in_sizes, int n_in,
                              void* d_out, int out_size, void* d_ws,
                              size_t ws_size, hipStream_t stream) {
  const float* x   = (const float*)d_in[0];
  const float* Wmu = (const float*)d_in[1];
  const float* Wls = (const float*)d_in[2];
  const float* bmu = (const float*)d_in[3];
  const float* bls = (const float*)d_in[4];
  const float* e   = (const float*)d_in[5];
  const float* eb  = (const float*)d_in[6];
  float* out = (float*)d_out;

  float* s  = (float*)d_ws;          // D*O floats = 512 KB
  float* sb = s + D_ * O_;           // O floats

  // 1) sigma precompute (tiny)
  sigma_kernel<<<(D_ * O_) / 256, 256, 0, stream>>>(Wls, bls, s, sb);
  // 2) mean-path GEMM via WMMA: 512 tiles, 8 waves/block -> 64 blocks
  gemm_wmma_kernel<<<64, 256, 0, stream>>>(x, Wmu, out);
  // 3) 256 MB e-stream + bias, accumulate into out
  noise_bias_kernel<<<(B_ * O4_) / 256, 256, 0, stream>>>(x, s, bmu, sb, e, eb,
                                                          out);
}